// MuonLonghornVisionMixer_57964878627185
// MI455X (gfx1250) — compile-verified
//
#include <hip/hip_runtime.h>
#include <hip/hip_bf16.h>
#include <math.h>

typedef float v2f __attribute__((ext_vector_type(2)));
typedef float v8f __attribute__((ext_vector_type(8)));

#define B_SZ   8
#define L_SZ   2048
#define DMODEL 512
#define DINNER 1024
#define DSSM   512
#define DSTATE 16
#define DTRANK 32
#define NPROJ  576   // 544 real channels (16 k + 16 q + 512 dt), padded to 9*64
#define TC     64

__device__ __forceinline__ v8f wmma_f32(v2f a, v2f b, v8f c) {
    // D = A(16x4, f32) * B(4x16, f32) + C(16x16, f32)
    return __builtin_amdgcn_wmma_f32_16x16x4_f32(false, a, false, b, (short)0, c, false, false);
}

// ---------------------------------------------------------------------------
// Build W_cat (576 x 512):
//   rows 0..15   = W_xproj rows 32..47  (k)
//   rows 16..31  = W_xproj rows 48..63  (q)
//   rows 32..543 = W_dt @ W_xproj[:32]  (effective dt projection, 512x512)
//   rows 544..575 = 0 (padding)
// ---------------------------------------------------------------------------
__global__ void k_wcat(const float* __restrict__ Wx, const float* __restrict__ Wdt,
                       float* __restrict__ Wcat) {
    int idx = blockIdx.x * blockDim.x + threadIdx.x;   // over 576*512
    int c = idx & (DSSM - 1);
    int e = idx >> 9;
    float v;
    if (e < 32) {
        v = Wx[(32 + e) * DSSM + c];
    } else if (e < 544) {
        int d = e - 32;
        float s = 0.f;
        #pragma unroll
        for (int r = 0; r < DTRANK; ++r)
            s += Wdt[d * DTRANK + r] * Wx[r * DSSM + c];
        v = s;
    } else {
        v = 0.f;
    }
    Wcat[(size_t)e * DSSM + c] = v;
}

// ---------------------------------------------------------------------------
// Input projection: xzT[b][e][l] = sum_d H[b][l][d] * Win[e][d]
// M = B*L (rows of (b,l)), N = DINNER, K = DMODEL. Transposed (channel-major)
// store. Wave: one 16-row M tile x 4 N tiles.
// ---------------------------------------------------------------------------
__global__ void __launch_bounds__(256) k_gemm_in(const float* __restrict__ H,
                                                 const float* __restrict__ Win,
                                                 float* __restrict__ xzT) {
    const int lane = threadIdx.x & 31;
    const int wave = threadIdx.x >> 5;
    const int mt = blockIdx.y * 8 + wave;
    const int row0 = mt << 4;
    const int b  = row0 >> 11;
    const int l0 = row0 & (L_SZ - 1);
    const int n0 = blockIdx.x << 6;
    const int ml = lane & 15;
    const int ko = (lane >> 4) << 1;

    const float* Ap  = H   + (size_t)(row0 + ml) * DMODEL + ko;
    const float* Bp  = Win + (size_t)(n0 + ml) * DMODEL + ko;

    v8f acc[4] = {v8f{}, v8f{}, v8f{}, v8f{}};
    #pragma unroll 4
    for (int k = 0; k < DMODEL; k += 4) {
        v2f a = *(const v2f*)(Ap + k);
        v2f b0 = *(const v2f*)(Bp + k);
        v2f b1 = *(const v2f*)(Bp + 16 * DMODEL + k);
        v2f b2 = *(const v2f*)(Bp + 32 * DMODEL + k);
        v2f b3 = *(const v2f*)(Bp + 48 * DMODEL + k);
        acc[0] = wmma_f32(a, b0, acc[0]);
        acc[1] = wmma_f32(a, b1, acc[1]);
        acc[2] = wmma_f32(a, b2, acc[2]);
        acc[3] = wmma_f32(a, b3, acc[3]);
    }
    const int lhalf = (lane >> 4) << 3;   // rows 0..7 or 8..15 of the tile
    #pragma unroll
    for (int j = 0; j < 4; ++j) {
        int e = n0 + j * 16 + ml;
        float* op = xzT + ((size_t)b * DINNER + e) * L_SZ + l0 + lhalf;
        *(float4*)(op)     = make_float4(acc[j][0], acc[j][1], acc[j][2], acc[j][3]);
        *(float4*)(op + 4) = make_float4(acc[j][4], acc[j][5], acc[j][6], acc[j][7]);
    }
}

// ---------------------------------------------------------------------------
// Depthwise conv (K=4, pad (1,2)) + SiLU.  x half -> xconv, z half -> ycat[512:]
// ---------------------------------------------------------------------------
__global__ void k_conv(const float* __restrict__ xzT,
                       const float* __restrict__ cxw, const float* __restrict__ cxb,
                       const float* __restrict__ czw, const float* __restrict__ czb,
                       float* __restrict__ xconv, float* __restrict__ ycat) {
    size_t idx = (size_t)blockIdx.x * blockDim.x + threadIdx.x;  // B*DINNER*L
    int l  = (int)(idx & (L_SZ - 1));
    int ch = (int)((idx >> 11) & (DINNER - 1));
    int b  = (int)(idx >> 21);
    const float* in = xzT + (idx - (size_t)l);
    float w0, w1, w2, w3, bias;
    if (ch < DSSM) {
        w0 = cxw[ch * 4 + 0]; w1 = cxw[ch * 4 + 1];
        w2 = cxw[ch * 4 + 2]; w3 = cxw[ch * 4 + 3];
        bias = cxb[ch];
    } else {
        int c = ch - DSSM;
        w0 = czw[c * 4 + 0]; w1 = czw[c * 4 + 1];
        w2 = czw[c * 4 + 2]; w3 = czw[c * 4 + 3];
        bias = czb[c];
    }
    float s = bias + w1 * in[l];
    if (l - 1 >= 0)   s += w0 * in[l - 1];
    if (l + 1 < L_SZ) s += w2 * in[l + 1];
    if (l + 2 < L_SZ) s += w3 * in[l + 2];
    float y = s / (1.f + __expf(-s));   // SiLU
    if (ch < DSSM)
        xconv[((size_t)b * DSSM + ch) * L_SZ + l] = y;
    else
        ycat[((size_t)b * DINNER + ch) * L_SZ + l] = y;
}

// ---------------------------------------------------------------------------
// Fused projection: proj[b][e][l] = sum_d Wcat[e][d] * xconv[b][d][l]
// e<16: k, e in 16..31: q, e in 32..543: raw dt (pre-bias, pre-softplus).
// A is read strided (channel-major activations). Transposed store.
// ---------------------------------------------------------------------------
__global__ void __launch_bounds__(256) k_gemm_proj(const float* __restrict__ Xc,
                                                   const float* __restrict__ Wcat,
                                                   float* __restrict__ proj) {
    const int lane = threadIdx.x & 31;
    const int wave = threadIdx.x >> 5;
    const int mt = blockIdx.y * 8 + wave;
    const int row0 = mt << 4;
    const int b  = row0 >> 11;
    const int l0 = row0 & (L_SZ - 1);
    const int n0 = blockIdx.x << 6;
    const int ml = lane & 15;
    const int ko = (lane >> 4) << 1;

    const float* Ap = Xc + (size_t)b * DSSM * L_SZ + l0 + ml;   // + k*L_SZ per k
    const float* Bp = Wcat + (size_t)(n0 + ml) * DSSM + ko;

    v8f acc[4] = {v8f{}, v8f{}, v8f{}, v8f{}};
    for (int k = 0; k < DSSM; k += 4) {
        v2f a;
        a.x = Ap[(size_t)(k + ko) * L_SZ];
        a.y = Ap[(size_t)(k + ko + 1) * L_SZ];
        v2f b0 = *(const v2f*)(Bp + k);
        v2f b1 = *(const v2f*)(Bp + 16 * DSSM + k);
        v2f b2 = *(const v2f*)(Bp + 32 * DSSM + k);
        v2f b3 = *(const v2f*)(Bp + 48 * DSSM + k);
        acc[0] = wmma_f32(a, b0, acc[0]);
        acc[1] = wmma_f32(a, b1, acc[1]);
        acc[2] = wmma_f32(a, b2, acc[2]);
        acc[3] = wmma_f32(a, b3, acc[3]);
    }
    const int lhalf = (lane >> 4) << 3;
    #pragma unroll
    for (int j = 0; j < 4; ++j) {
        int e = n0 + j * 16 + ml;
        float* op = proj + ((size_t)b * NPROJ + e) * L_SZ + l0 + lhalf;
        *(float4*)(op)     = make_float4(acc[j][0], acc[j][1], acc[j][2], acc[j][3]);
        *(float4*)(op + 4) = make_float4(acc[j][4], acc[j][5], acc[j][6], acc[j][7]);
    }
}

// ---------------------------------------------------------------------------
// Longhorn scan. One lane per (b,d) channel; 16-entry state in VGPRs.
// k/q tiles staged LDS-direct with GLOBAL_LOAD_ASYNC_TO_LDS_B128 (ASYNCcnt),
// overlapped with per-lane u/dt LDS staging. BETA=0, ALPHA=1 =>
//   S += eps*(u - S*k)*k ; y = sum_n S*q + Dskip*u.
// ---------------------------------------------------------------------------
__global__ void __launch_bounds__(64) k_scan(const float* __restrict__ proj,
                                             const float* __restrict__ xconv,
                                             const float* __restrict__ bdt,
                                             const float* __restrict__ dskip,
                                             float* __restrict__ ycat) {
    __shared__ __align__(16) float kqs[32][TC];   // rows 0..15 = k, 16..31 = q
    __shared__ float k2s[TC];
    __shared__ float us[64][TC + 1];
    __shared__ float dts[64][TC + 1];

    const int tid = threadIdx.x;                 // 0..63
    const int b = blockIdx.x >> 3;
    const int d = ((blockIdx.x & 7) << 6) + tid;

    const float* kq    = proj  + (size_t)b * NPROJ * L_SZ;             // ch 0..31
    const float* dtrow = proj  + ((size_t)b * NPROJ + 32 + d) * L_SZ;
    const float* urow  = xconv + ((size_t)b * DSSM + d) * L_SZ;
    float*       yrow  = ycat  + ((size_t)b * DINNER + d) * L_SZ;

    const float bias = bdt[d];
    const float dsk  = dskip[d];

    float S[DSTATE];
    #pragma unroll
    for (int n = 0; n < DSTATE; ++n) S[n] = 0.f;

    for (int t0 = 0; t0 < L_SZ; t0 += TC) {
        __syncthreads();                          // protect LDS from prev chunk

        // --- async stage k/q: 32 rows x TC floats, B128 direct-to-LDS -------
        #pragma unroll
        for (int p = 0; p < 8; ++p) {
            int g  = p * 64 + tid;                // 0..511 float4 groups
            int ch = g >> 4;                      // 0..31
            int t4 = (g & 15) << 2;               // 0,4,...,60
            const float* gp = kq + (size_t)ch * L_SZ + t0 + t4;
            unsigned lds = (unsigned)(size_t)&kqs[ch][t4];   // LDS byte offset
            asm volatile("global_load_async_to_lds_b128 %0, %1, off"
                         :: "v"(lds), "v"(gp) : "memory");
        }

        // --- per-lane u / dt chunk staging (contiguous b128 loads) ----------
        for (int tt = 0; tt < TC; tt += 4) {
            float4 uv = *(const float4*)(urow + t0 + tt);
            us[tid][tt] = uv.x; us[tid][tt + 1] = uv.y;
            us[tid][tt + 2] = uv.z; us[tid][tt + 3] = uv.w;
            float4 dv = *(const float4*)(dtrow + t0 + tt);
            dts[tid][tt] = dv.x; dts[tid][tt + 1] = dv.y;
            dts[tid][tt + 2] = dv.z; dts[tid][tt + 3] = dv.w;
        }

        asm volatile("s_wait_asynccnt 0x0" ::: "memory");   // k/q landed in LDS
        __syncthreads();

        {   // k2[t] = sum_n k[n][t]^2, one t per thread (TC == blockDim)
            float s = 0.f;
            #pragma unroll
            for (int n = 0; n < DSTATE; ++n) { float kv = kqs[n][tid]; s += kv * kv; }
            k2s[tid] = s;
        }
        __syncthreads();

        if (t0 + TC < L_SZ) {                     // global_prefetch for next chunk
            __builtin_prefetch(urow + t0 + TC, 0, 1);
            __builtin_prefetch(dtrow + t0 + TC, 0, 1);
        }

        for (int t = 0; t < TC; ++t) {
            float u  = us[tid][t];
            float dv = dts[tid][t] + bias;
            float sp = (dv > 20.f) ? dv : __logf(1.f + __expf(dv));  // softplus
            float eps = sp / (1.f + sp * k2s[t]);
            float y = 0.f;
            #pragma unroll
            for (int n = 0; n < DSTATE; ++n) {
                float kv = kqs[n][t];
                S[n] += eps * (u - S[n] * kv) * kv;
                y += S[n] * kqs[DSTATE + n][t];
            }
            yrow[t0 + t] = y + dsk * u;
        }
    }
}

// ---------------------------------------------------------------------------
// Output projection: out[b][l][d] = sum_e ycat[b][e][l] * Wout[d][e]
// M = B*L, N = DMODEL, K = DINNER. Row-major store.
// ---------------------------------------------------------------------------
__global__ void __launch_bounds__(256) k_gemm_out(const float* __restrict__ Yc,
                                                  const float* __restrict__ Wout,
                                                  float* __restrict__ out) {
    const int lane = threadIdx.x & 31;
    const int wave = threadIdx.x >> 5;
    const int mt = blockIdx.y * 8 + wave;
    const int row0 = mt << 4;
    const int b  = row0 >> 11;
    const int l0 = row0 & (L_SZ - 1);
    const int n0 = blockIdx.x << 6;
    const int ml = lane & 15;
    const int ko = (lane >> 4) << 1;

    const float* Ap = Yc + (size_t)b * DINNER * L_SZ + l0 + ml;    // + k*L_SZ
    const float* Bp = Wout + (size_t)(n0 + ml) * DINNER + ko;

    v8f acc[4] = {v8f{}, v8f{}, v8f{}, v8f{}};
    for (int k = 0; k < DINNER; k += 4) {
        v2f a;
        a.x = Ap[(size_t)(k + ko) * L_SZ];
        a.y = Ap[(size_t)(k + ko + 1) * L_SZ];
        v2f b0 = *(const v2f*)(Bp + k);
        v2f b1 = *(const v2f*)(Bp + 16 * DINNER + k);
        v2f b2 = *(const v2f*)(Bp + 32 * DINNER + k);
        v2f b3 = *(const v2f*)(Bp + 48 * DINNER + k);
        acc[0] = wmma_f32(a, b0, acc[0]);
        acc[1] = wmma_f32(a, b1, acc[1]);
        acc[2] = wmma_f32(a, b2, acc[2]);
        acc[3] = wmma_f32(a, b3, acc[3]);
    }
    const int lhalf = (lane >> 4) << 3;
    #pragma unroll
    for (int j = 0; j < 4; ++j) {
        int col = n0 + j * 16 + ml;
        #pragma unroll
        for (int r = 0; r < 8; ++r) {
            out[(size_t)(row0 + lhalf + r) * DMODEL + col] = acc[j][r];
        }
    }
}

// ---------------------------------------------------------------------------
extern "C" void kernel_launch(void* const* d_in, const int* in_sizes, int n_in,
                              void* d_out, int out_size, void* d_ws, size_t ws_size,
                              hipStream_t stream) {
    (void)in_sizes; (void)n_in; (void)out_size; (void)ws_size;
    const float* H    = (const float*)d_in[0];   // (B, L, DMODEL)
    const float* Win  = (const float*)d_in[1];   // (DINNER, DMODEL)
    const float* cxw  = (const float*)d_in[2];   // (DSSM, 1, 4)
    const float* cxb  = (const float*)d_in[3];   // (DSSM,)
    const float* czw  = (const float*)d_in[4];   // (DSSM, 1, 4)
    const float* czb  = (const float*)d_in[5];   // (DSSM,)
    const float* Wxp  = (const float*)d_in[6];   // (64, DSSM)
    const float* Wdt  = (const float*)d_in[7];   // (DSSM, 32)
    const float* bdt  = (const float*)d_in[8];   // (DSSM,)
    const float* Dsk  = (const float*)d_in[9];   // (DSSM,)
    const float* Wout = (const float*)d_in[10];  // (DMODEL, DINNER)
    float* out = (float*)d_out;

    char* ws = (char*)d_ws;
    // workspace layout (proj aliases xzT, which is dead after the conv):
    float* xzT   = (float*)(ws + 0);                 //  8*1024*2048*4 = 64 MiB
    float* proj  = (float*)(ws + 0);                 //  8* 576*2048*4 = 36 MiB (alias)
    float* xconv = (float*)(ws + 67108864ull);       //  8* 512*2048*4 = 32 MiB
    float* ycat  = (float*)(ws + 100663296ull);      //  8*1024*2048*4 = 64 MiB
    float* wcat  = (float*)(ws + 167772160ull);      //  576*512*4     = 1.125 MiB

    k_wcat<<<(NPROJ * DSSM) / 256, 256, 0, stream>>>(Wxp, Wdt, wcat);
    k_gemm_in<<<dim3(DINNER / 64, (B_SZ * L_SZ) / 128), 256, 0, stream>>>(H, Win, xzT);
    k_conv<<<(B_SZ * DINNER * L_SZ) / 256, 256, 0, stream>>>(xzT, cxw, cxb, czw, czb,
                                                             xconv, ycat);
    k_gemm_proj<<<dim3(NPROJ / 64, (B_SZ * L_SZ) / 128), 256, 0, stream>>>(xconv, wcat, proj);
    k_scan<<<B_SZ * (DSSM / 64), 64, 0, stream>>>(proj, xconv, bdt, Dsk, ycat);
    k_gemm_out<<<dim3(DMODEL / 64, (B_SZ * L_SZ) / 128), 256, 0, stream>>>(ycat, Wout, out);
}